// BasicConvolutionBlock_90615220011114
// MI455X (gfx1250) — compile-verified
//
#include <hip/hip_runtime.h>

typedef __attribute__((ext_vector_type(2))) float v2f;
typedef __attribute__((ext_vector_type(8))) float v8f;

#define NPTS      60000
#define KOFF      27
#define INC       64
#define OUTC      64
#define BN_EPS    1e-5f
#define WT_STRIDE 66   // padded LDS row stride (floats): keeps 8B alignment, spreads banks

// ws layout (floats): [0,64) channel sums, [64,128) channel sum-of-squares.

__global__ void zero_stats_kernel(float* __restrict__ stats) {
    if (threadIdx.x < 128) stats[threadIdx.x] = 0.0f;
}

// One wave = 16 voxels x 64 output channels via V_WMMA_F32_16X16X4_F32.
// 8 waves per block (128 voxels). k-offset is the OUTER loop: the block
// cooperatively stages W[k]^T (16.5 KB) in LDS once per k, so B fragments
// come from one ds_load_b64 each instead of 8 strided global b32 loads.
__global__ __launch_bounds__(256)
void spconv_gemm_wmma(const float* __restrict__ feats,
                      const int*   __restrict__ nbr_idx,
                      const int*   __restrict__ nbr_mask,
                      const float* __restrict__ W,
                      float*       __restrict__ y,
                      float*       __restrict__ stats) {
    __shared__ float wt[OUTC * WT_STRIDE];

    const int wave  = threadIdx.x >> 5;
    const int lane  = threadIdx.x & 31;
    const int tile  = blockIdx.x * 8 + wave;
    const bool active = tile < (NPTS / 16);      // wave-uniform predicate
    const int n0   = (active ? tile : 0) * 16;   // inactive waves redo tile 0
    const int row  = lane & 15;                  // A: M row / B,C,D: N column
    const int half = lane >> 4;                  // K pair (A/B) and M half (C/D)
    const int nrow = n0 + row;

    v8f c0 = {}, c1 = {}, c2 = {}, c3 = {};

    for (int k = 0; k < KOFF; ++k) {
        // ---- cooperative transpose-stage of W[k] into LDS (coalesced b128 reads)
        const float* wk = W + k * (INC * OUTC);
#pragma unroll
        for (int j = 0; j < 4; ++j) {
            const int e4 = (threadIdx.x + j * 256) * 4;   // covers all 4096 elems
            const float4 w4 = *(const float4*)(wk + e4);
            const int i = e4 >> 6;      // input channel
            const int o = e4 & 63;      // output channel (multiple of 4)
            wt[(o + 0) * WT_STRIDE + i] = w4.x;
            wt[(o + 1) * WT_STRIDE + i] = w4.y;
            wt[(o + 2) * WT_STRIDE + i] = w4.z;
            wt[(o + 3) * WT_STRIDE + i] = w4.w;
        }
        __syncthreads();

        // ---- per-lane gather row for this k-offset
        const int   idx = nbr_idx[nrow * KOFF + k];
        const float m   = (float)nbr_mask[nrow * KOFF + k];
        const float* frow = feats + idx * INC;

#pragma unroll 4
        for (int ks = 0; ks < INC / 4; ++ks) {
            const int kk = ks * 4 + half * 2;             // even -> 8B aligned
            // A fragment: contiguous float2 of gathered features, mask applied
            v2f a = *(const v2f*)(frow + kk);
            a *= m;
            // B fragments: wt[col][kk..kk+1] contiguous -> ds_load_b64
            v2f b0 = *(const v2f*)(&wt[(row +  0) * WT_STRIDE + kk]);
            v2f b1 = *(const v2f*)(&wt[(row + 16) * WT_STRIDE + kk]);
            v2f b2 = *(const v2f*)(&wt[(row + 32) * WT_STRIDE + kk]);
            v2f b3 = *(const v2f*)(&wt[(row + 48) * WT_STRIDE + kk]);
            c0 = __builtin_amdgcn_wmma_f32_16x16x4_f32(false, a, false, b0, (short)0, c0, false, false);
            c1 = __builtin_amdgcn_wmma_f32_16x16x4_f32(false, a, false, b1, (short)0, c1, false, false);
            c2 = __builtin_amdgcn_wmma_f32_16x16x4_f32(false, a, false, b2, (short)0, c2, false, false);
            c3 = __builtin_amdgcn_wmma_f32_16x16x4_f32(false, a, false, b3, (short)0, c3, false, false);
        }
        __syncthreads();   // protect wt before next k overwrites it
    }

    if (!active) return;

    // Store y tiles + fused BN statistics (sum / sumsq per channel).
    // C/D layout: VGPR v, lanes 0-15 -> M=v, lanes 16-31 -> M=v+8; N = lane&15.
#define EMIT_TILE(C, T)                                                        \
    do {                                                                       \
        float s = 0.0f, s2 = 0.0f;                                             \
        _Pragma("unroll")                                                      \
        for (int v = 0; v < 8; ++v) {                                          \
            const float val = (C)[v];                                          \
            y[(n0 + v + half * 8) * OUTC + (T) * 16 + row] = val;              \
            s  += val;                                                         \
            s2 += val * val;                                                   \
        }                                                                      \
        s  += __shfl_xor(s, 16);                                               \
        s2 += __shfl_xor(s2, 16);                                              \
        if (half == 0) {                                                       \
            atomicAdd(&stats[(T) * 16 + row], s);                              \
            atomicAdd(&stats[64 + (T) * 16 + row], s2);                        \
        }                                                                      \
    } while (0)

    EMIT_TILE(c0, 0);
    EMIT_TILE(c1, 1);
    EMIT_TILE(c2, 2);
    EMIT_TILE(c3, 3);
#undef EMIT_TILE
}

// In-place BatchNorm (batch statistics, biased variance) + ReLU over y.
__global__ __launch_bounds__(256)
void bn_relu_kernel(float* __restrict__ y,
                    const float* __restrict__ stats,
                    const float* __restrict__ gamma,
                    const float* __restrict__ beta) {
    const int i = blockIdx.x * blockDim.x + threadIdx.x;   // one float4 each
    const int total4 = NPTS * OUTC / 4;
    if (i >= total4) return;

    float4 v = ((const float4*)y)[i];
    float in[4] = { v.x, v.y, v.z, v.w };
    const int c0 = (i * 4) & (OUTC - 1);
    const float invN = 1.0f / (float)NPTS;

    float o[4];
#pragma unroll
    for (int j = 0; j < 4; ++j) {
        const int c = c0 + j;
        const float mean   = stats[c] * invN;
        const float var    = stats[64 + c] * invN - mean * mean;
        const float invstd = rsqrtf(var + BN_EPS);
        const float t = (in[j] - mean) * invstd * gamma[c] + beta[c];
        o[j] = fmaxf(t, 0.0f);
    }
    ((float4*)y)[i] = make_float4(o[0], o[1], o[2], o[3]);
}

extern "C" void kernel_launch(void* const* d_in, const int* in_sizes, int n_in,
                              void* d_out, int out_size, void* d_ws, size_t ws_size,
                              hipStream_t stream) {
    const float* feats    = (const float*)d_in[0];
    const int*   nbr_idx  = (const int*)  d_in[1];
    const int*   nbr_mask = (const int*)  d_in[2];
    const float* W        = (const float*)d_in[3];
    const float* gamma    = (const float*)d_in[4];
    const float* beta     = (const float*)d_in[5];
    float* out   = (float*)d_out;
    float* stats = (float*)d_ws;

    (void)in_sizes; (void)n_in; (void)out_size; (void)ws_size;

    zero_stats_kernel<<<1, 128, 0, stream>>>(stats);

    const int tiles  = NPTS / 16;                 // 3750 (exact)
    const int blocks = (tiles + 7) / 8;           // 469, 8 waves per block
    spconv_gemm_wmma<<<blocks, 256, 0, stream>>>(feats, nbr_idx, nbr_mask, W, out, stats);

    const int total4 = NPTS * OUTC / 4;           // 960000
    bn_relu_kernel<<<(total4 + 255) / 256, 256, 0, stream>>>(out, stats, gamma, beta);
}